// LiquidLag1_67078799229347
// MI455X (gfx1250) — compile-verified
//
#include <hip/hip_runtime.h>
#include <cstdint>

#define LOG2E 1.442695040888963f
#define EPSV  1e-8f

// ---------------- d_ws layout (float offsets) ----------------
#define OFF_P1   0        // 64x64 float4  (layer1 recurrent {sgl,ofs,e,a})
#define OFF_PS1  16384    // 16x64 float4  (layer1 sensory)
#define OFF_P2   20480    // 32x32 float4  (layer2 recurrent)
#define OFF_PS2  24576    // 32x32 float4  (layer2 sensory)
#define NPARAM4  28672    // floats copied to LDS via TDM
#define OFF_GL1  28672
#define OFF_GVL1 28736
#define OFF_CMT1 28800
#define OFF_GL2  28864
#define OFF_GVL2 28896
#define OFF_CMT2 28928
#define OFF_IW1  28960
#define OFF_IB1  28976
#define OFF_K1   28992
#define OFF_K2   29024
#define OFF_OW2  29056
#define OFF_OB2  29072
#define OFF_FCW  29088
#define OFF_FCB  29104
#define OFF_H    29120    // 512x16 final hidden
#define WAVES_PER_BLOCK 4
#define SCR_PER_WAVE 160  // v1(64)+v2(32)+xi1(16)+xi2(32) padded
#define LDS_FLOATS (NPARAM4 + WAVES_PER_BLOCK * SCR_PER_WAVE)
#define LDS_BYTES  (LDS_FLOATS * 4)

typedef unsigned int u32x4 __attribute__((ext_vector_type(4)));
typedef int          i32x4 __attribute__((ext_vector_type(4)));
typedef int          i32x8 __attribute__((ext_vector_type(8)));
typedef float        v2f   __attribute__((ext_vector_type(2)));
typedef float        v8f   __attribute__((ext_vector_type(8)));

struct LtcIns {
  const float *x;
  const float *iw1, *ib1, *smu1, *ssg1, *sw1, *mu1, *sg1, *w1;
  const float *gleak1, *vleak1, *cm1, *ow1, *ob1, *serev1, *erev1;
  const int   *msen1, *mrec1;
  const float *iw2, *ib2, *smu2, *ssg2, *sw2, *mu2, *sg2, *w2;
  const float *gleak2, *vleak2, *cm2, *ow2, *ob2, *serev2, *erev2;
  const int   *msen2, *mrec2;
  const float *fcw, *fcb;
};

__device__ __forceinline__ float softplusf(float v) {
  return (v > 20.f) ? v : log1pf(__expf(v));
}
// sigmoid(sg*(v-mu)) with prefolded sgl=-log2e*sg, ofs=log2e*sg*mu
__device__ __forceinline__ float gatef(float sgl, float ofs, float v) {
  return __builtin_amdgcn_rcpf(1.0f + __builtin_amdgcn_exp2f(fmaf(sgl, v, ofs)));
}
__device__ __forceinline__ float sigmoidf_(float t) {
  return __builtin_amdgcn_rcpf(1.0f + __builtin_amdgcn_exp2f(-LOG2E * t));
}
__device__ __forceinline__ void wave_sync() {
  __builtin_amdgcn_fence(__ATOMIC_ACQ_REL, "wavefront");
  __builtin_amdgcn_wave_barrier();
}

// ------------- prep: fold softplus/masks/erev into packed float4 records -------------
__global__ __launch_bounds__(256) void ltc_prep(LtcIns in, float* __restrict__ ws) {
  const int tid = threadIdx.x;
  for (int i = tid; i < 4096; i += 256) {  // layer1 recurrent 64x64
    float a  = softplusf(in.w1[i]) * (float)in.mrec1[i];
    float sg = in.sg1[i], mu = in.mu1[i];
    float4 o; o.x = -LOG2E * sg; o.y = LOG2E * sg * mu; o.z = a * in.erev1[i]; o.w = a;
    reinterpret_cast<float4*>(ws + OFF_P1)[i] = o;
  }
  for (int i = tid; i < 1024; i += 256) {  // layer1 sensory 16x64
    float a  = softplusf(in.sw1[i]) * (float)in.msen1[i];
    float sg = in.ssg1[i], mu = in.smu1[i];
    float4 o; o.x = -LOG2E * sg; o.y = LOG2E * sg * mu; o.z = a * in.serev1[i]; o.w = a;
    reinterpret_cast<float4*>(ws + OFF_PS1)[i] = o;
  }
  for (int i = tid; i < 1024; i += 256) {  // layer2 recurrent 32x32
    float a  = softplusf(in.w2[i]) * (float)in.mrec2[i];
    float sg = in.sg2[i], mu = in.mu2[i];
    float4 o; o.x = -LOG2E * sg; o.y = LOG2E * sg * mu; o.z = a * in.erev2[i]; o.w = a;
    reinterpret_cast<float4*>(ws + OFF_P2)[i] = o;
  }
  for (int i = tid; i < 1024; i += 256) {  // layer2 sensory 32x32
    float a  = softplusf(in.sw2[i]) * (float)in.msen2[i];
    float sg = in.ssg2[i], mu = in.smu2[i];
    float4 o; o.x = -LOG2E * sg; o.y = LOG2E * sg * mu; o.z = a * in.serev2[i]; o.w = a;
    reinterpret_cast<float4*>(ws + OFF_PS2)[i] = o;
  }
  if (tid < 64) {
    float gl = softplusf(in.gleak1[tid]);
    ws[OFF_GL1 + tid]  = gl;
    ws[OFF_GVL1 + tid] = gl * in.vleak1[tid];
    ws[OFF_CMT1 + tid] = softplusf(in.cm1[tid]) * 6.0f;
  }
  if (tid < 32) {
    float gl = softplusf(in.gleak2[tid]);
    ws[OFF_GL2 + tid]  = gl;
    ws[OFF_GVL2 + tid] = gl * in.vleak2[tid];
    ws[OFF_CMT2 + tid] = softplusf(in.cm2[tid]) * 6.0f;
    ws[OFF_K1 + tid]   = in.ow1[tid] * in.iw2[tid];                 // fold layer1 out -> layer2 in
    ws[OFF_K2 + tid]   = fmaf(in.ob1[tid], in.iw2[tid], in.ib2[tid]);
  }
  if (tid < 16) {
    ws[OFF_IW1 + tid] = in.iw1[tid];
    ws[OFF_IB1 + tid] = in.ib1[tid];
    ws[OFF_OW2 + tid] = in.ow2[tid];
    ws[OFF_OB2 + tid] = in.ob2[tid];
    ws[OFF_FCW + tid] = in.fcw[tid];
  }
  if (tid == 0) ws[OFF_FCB] = in.fcb[0];
}

// ------------- main fused 2-layer LTC scan: one wave per batch sequence -------------
__global__ __launch_bounds__(128) void ltc_main(const float* __restrict__ x,
                                                const float* __restrict__ ws,
                                                float* __restrict__ hout) {
  extern __shared__ float lds[];
  const int tid  = threadIdx.x;
  const int lane = tid & 31;
  const int wv   = tid >> 5;
  const int b    = blockIdx.x * WAVES_PER_BLOCK + wv;

  // ---- stage 112KB param block into LDS with the Tensor Data Mover ----
  if (tid == 0) {
    uint64_t ga = (uint64_t)(uintptr_t)ws;       // tile start in global
    uint32_t la = (uint32_t)(uintptr_t)lds;      // flat->LDS: low 32 bits are LDS byte addr
    u32x4 g0;
    g0.x = 1u;                                   // count=1, load, no gather
    g0.y = la;                                   // lds_addr
    g0.z = (uint32_t)(ga & 0xffffffffu);         // global_addr[31:0]
    g0.w = (uint32_t)((ga >> 32) & 0x01ffffffu) | (2u << 30);  // addr[56:32] | type=2
    i32x8 g1;
    g1[0] = 0x00020000;                               // data_size = 4B
    g1[1] = (int)((NPARAM4 & 0xffffu) << 16);         // tensor_dim0 lo16
    g1[2] = (int)(((NPARAM4 >> 16) & 0xffffu) | (1u << 16)); // dim0 hi16 | tensor_dim1=1
    g1[3] = (int)((NPARAM4 & 0xffffu) << 16);         // tile_dim0 = NPARAM4 elements
    g1[4] = 1;                                        // tile_dim1 = 1
    g1[5] = NPARAM4;                                  // tensor_dim0_stride lo32
    g1[6] = (int)((NPARAM4 & 0xffffu) << 16);         // dim1_stride lo16
    g1[7] = 0;
    i32x4 z4 = {0, 0, 0, 0};
#if __has_include(<hip/amd_detail/amd_gfx1250_TDM.h>)
    i32x8 z8 = {0, 0, 0, 0, 0, 0, 0, 0};
    __builtin_amdgcn_tensor_load_to_lds(g0, g1, z4, z4, z8, 0);
#else
    __builtin_amdgcn_tensor_load_to_lds(g0, g1, z4, z4, 0);
#endif
  }
  __builtin_amdgcn_s_wait_tensorcnt(0);
  __syncthreads();

  // per-lane constants (lane owns layer1 columns lane, lane+32; layer2 column lane)
  const int jA = lane, jB = lane + 32;
  const float cmtA = ws[OFF_CMT1 + jA], glA = ws[OFF_GL1 + jA], gvlA = ws[OFF_GVL1 + jA];
  const float cmtB = ws[OFF_CMT1 + jB], glB = ws[OFF_GL1 + jB], gvlB = ws[OFF_GVL1 + jB];
  const float cmt2 = ws[OFF_CMT2 + lane], gl2 = ws[OFF_GL2 + lane], gvl2 = ws[OFF_GVL2 + lane];
  const float k1 = ws[OFF_K1 + lane], k2 = ws[OFF_K2 + lane];
  const int l16 = lane & 15;
  const float iw1 = ws[OFF_IW1 + l16], ib1 = ws[OFF_IB1 + l16];
  const float ow2 = ws[OFF_OW2 + l16], ob2 = ws[OFF_OB2 + l16];

  const float4* __restrict__ P1v  = reinterpret_cast<const float4*>(lds);
  const float4* __restrict__ PS1v = reinterpret_cast<const float4*>(lds + OFF_PS1);
  const float4* __restrict__ P2v  = reinterpret_cast<const float4*>(lds + OFF_P2);
  const float4* __restrict__ PS2v = reinterpret_cast<const float4*>(lds + OFF_PS2);
  float* wscr = lds + NPARAM4 + wv * SCR_PER_WAVE;
  float* sv1 = wscr;        // 64: layer1 state, broadcast via LDS
  float* sv2 = wscr + 64;   // 32: layer2 state
  float* sx1 = wscr + 96;   // 16: layer1 input xi
  float* sx2 = wscr + 112;  // 32: layer2 input xi

  float v1a = 0.f, v1b = 0.f, v2 = 0.f;
  const float* xb = x + (size_t)b * 1024 * 8;

  for (int t = 0; t < 1024; ++t) {
    // layer1 input: [x_t (8) , x_{t-1} (8)] * input_w + input_b
    if (lane < 16) {
      float val = 0.f;
      if (lane < 8)      val = xb[t * 8 + lane];
      else if (t > 0)    val = xb[(t - 1) * 8 + lane - 8];
      sx1[lane] = fmaf(val, iw1, ib1);
    }
    wave_sync();

    float nsA = 0.f, dsA = 0.f, nsB = 0.f, dsB = 0.f;
#pragma unroll 4
    for (int s = 0; s < 16; ++s) {
      float xv = sx1[s];
      float4 pa = PS1v[s * 64 + jA];
      float4 pb = PS1v[s * 64 + jB];
      float ga = gatef(pa.x, pa.y, xv);
      float gb = gatef(pb.x, pb.y, xv);
      nsA = fmaf(pa.z, ga, nsA); dsA = fmaf(pa.w, ga, dsA);
      nsB = fmaf(pb.z, gb, nsB); dsB = fmaf(pb.w, gb, dsB);
    }
#pragma unroll 1
    for (int u = 0; u < 6; ++u) {          // layer1 ODE unfolds
      sv1[jA] = v1a; sv1[jB] = v1b;
      wave_sync();
      float nA = nsA, dA = dsA, nB = nsB, dB = dsB;
#pragma unroll 4
      for (int i = 0; i < 64; ++i) {
        float vi = sv1[i];                  // LDS broadcast
        float4 pa = P1v[i * 64 + jA];       // ds_load_b128
        float4 pb = P1v[i * 64 + jB];
        float ga = gatef(pa.x, pa.y, vi);
        float gb = gatef(pb.x, pb.y, vi);
        nA = fmaf(pa.z, ga, nA); dA = fmaf(pa.w, ga, dA);
        nB = fmaf(pb.z, gb, nB); dB = fmaf(pb.w, gb, dB);
      }
      v1a = (fmaf(cmtA, v1a, gvlA) + nA) * __builtin_amdgcn_rcpf(cmtA + glA + dA + EPSV);
      v1b = (fmaf(cmtB, v1b, gvlB) + nB) * __builtin_amdgcn_rcpf(cmtB + glB + dB + EPSV);
      wave_sync();
    }

    // layer2 input (layer1 out_w/out_b folded with layer2 input_w/input_b)
    sx2[lane] = fmaf(v1a, k1, k2);
    wave_sync();
    float ns2 = 0.f, ds2 = 0.f;
#pragma unroll 4
    for (int s = 0; s < 32; ++s) {
      float xv = sx2[s];
      float4 p = PS2v[s * 32 + lane];
      float g = gatef(p.x, p.y, xv);
      ns2 = fmaf(p.z, g, ns2); ds2 = fmaf(p.w, g, ds2);
    }
#pragma unroll 1
    for (int u = 0; u < 6; ++u) {          // layer2 ODE unfolds
      sv2[lane] = v2;
      wave_sync();
      float n = ns2, d = ds2;
#pragma unroll 8
      for (int i = 0; i < 32; ++i) {
        float vi = sv2[i];
        float4 p = P2v[i * 32 + lane];
        float g = gatef(p.x, p.y, vi);
        n = fmaf(p.z, g, n); d = fmaf(p.w, g, d);
      }
      v2 = (fmaf(cmt2, v2, gvl2) + n) * __builtin_amdgcn_rcpf(cmt2 + gl2 + d + EPSV);
      wave_sync();
    }
  }
  if (lane < 16) hout[b * 16 + lane] = fmaf(v2, ow2, ob2);
}

// ------------- final dense layer: sigmoid(h(512x16) @ fc_w(16x1) + b) via WMMA f32 -------------
__global__ __launch_bounds__(32) void ltc_fc_wmma(const float* __restrict__ h,
                                                  const float* __restrict__ fcw,
                                                  const float* __restrict__ fcb,
                                                  float* __restrict__ out) {
  const int lane = threadIdx.x;
  const int tile = blockIdx.x;           // 32 tiles of 16 rows
  const int m    = lane & 15;
  const int koff = (lane >> 4) << 1;     // A layout: VGPR0 = K{0,2}, VGPR1 = K{1,3}
  const float* hrow = h + (tile * 16 + m) * 16;
  v8f c = {0.f, 0.f, 0.f, 0.f, 0.f, 0.f, 0.f, 0.f};
#pragma unroll
  for (int kc = 0; kc < 4; ++kc) {       // K=16 in 4 chunks of 4
    v2f a, bb;
    a.x  = hrow[kc * 4 + koff];
    a.y  = hrow[kc * 4 + koff + 1];
    bb.x = fcw[kc * 4 + koff];           // replicate w across N; only column 0 consumed
    bb.y = fcw[kc * 4 + koff + 1];
    c = __builtin_amdgcn_wmma_f32_16x16x4_f32(false, a, false, bb, (short)0, c, false, false);
  }
  const float bias = fcb[0];
  // D layout: VGPR q -> M=q (lanes 0-15, N=lane) / M=q+8 (lanes 16-31); column 0 = lanes 0,16
  if (lane == 0) {
#pragma unroll
    for (int q = 0; q < 8; ++q) out[tile * 16 + q] = sigmoidf_(c[q] + bias);
  } else if (lane == 16) {
#pragma unroll
    for (int q = 0; q < 8; ++q) out[tile * 16 + 8 + q] = sigmoidf_(c[q] + bias);
  }
}

extern "C" void kernel_launch(void* const* d_in, const int* in_sizes, int n_in,
                              void* d_out, int out_size, void* d_ws, size_t ws_size,
                              hipStream_t stream) {
  (void)in_sizes; (void)n_in; (void)out_size; (void)ws_size;
  const float* const* f = (const float* const*)d_in;
  LtcIns in;
  in.x = f[0];
  in.iw1 = f[1];  in.ib1 = f[2];  in.smu1 = f[3];  in.ssg1 = f[4];  in.sw1 = f[5];
  in.mu1 = f[6];  in.sg1 = f[7];  in.w1 = f[8];
  in.gleak1 = f[9];  in.vleak1 = f[10]; in.cm1 = f[11];
  in.ow1 = f[12]; in.ob1 = f[13]; in.serev1 = f[14]; in.erev1 = f[15];
  in.msen1 = (const int*)d_in[16]; in.mrec1 = (const int*)d_in[17];
  in.iw2 = f[18]; in.ib2 = f[19]; in.smu2 = f[20]; in.ssg2 = f[21]; in.sw2 = f[22];
  in.mu2 = f[23]; in.sg2 = f[24]; in.w2 = f[25];
  in.gleak2 = f[26]; in.vleak2 = f[27]; in.cm2 = f[28];
  in.ow2 = f[29]; in.ob2 = f[30]; in.serev2 = f[31]; in.erev2 = f[32];
  in.msen2 = (const int*)d_in[33]; in.mrec2 = (const int*)d_in[34];
  in.fcw = f[35]; in.fcb = f[36];

  float* ws = (float*)d_ws;
  hipFuncSetAttribute(reinterpret_cast<const void*>(&ltc_main),
                      hipFuncAttributeMaxDynamicSharedMemorySize, LDS_BYTES);
  ltc_prep<<<1, 256, 0, stream>>>(in, ws);
  ltc_main<<<512 / WAVES_PER_BLOCK, WAVES_PER_BLOCK * 32, LDS_BYTES, stream>>>(in.x, ws, ws + OFF_H);
  ltc_fc_wmma<<<32, 32, 0, stream>>>(ws + OFF_H, ws + OFF_FCW, ws + OFF_FCB, (float*)d_out);
}